// GNN_EBM_Layer_edge_51874615001421
// MI455X (gfx1250) — compile-verified
//
#include <hip/hip_runtime.h>

// Problem constants from the reference: B=2, T=10000, M=100000, D=64.
typedef float v2f __attribute__((ext_vector_type(2)));
typedef float v8f __attribute__((ext_vector_type(8)));

static constexpr int kB = 2;
static constexpr int kT = 10000;
static constexpr int kM = 100000;
static constexpr int kD = 64;
static constexpr int WS = 68;  // padded LDS row stride (floats) -> conflict-free stride-4 bank pattern
static constexpr int R  = 2;   // 16-row tiles per wave: B-fragments reused R times -> halves LDS W traffic

__global__ __launch_bounds__(128) void gnn_fused_wmma_f32(
    const float* __restrict__ x1,   // (B, T, 2, D)
    const float* __restrict__ x2,   // (B, M, 4, D)
    const float* __restrict__ Wm,   // (D, D)
    const float* __restrict__ bm,   // (D)
    const float* __restrict__ We,   // (D, D)
    const float* __restrict__ be,   // (D)
    const int*   __restrict__ edge, // (2, M)
    float*       __restrict__ out)  // (B, M, 4, D)
{
    __shared__ float sWm[64 * WS];
    __shared__ float sWe[64 * WS];
    __shared__ float sH[4][16 * R * WS];  // per-wave h tiles (C-layout -> A-layout transpose)

    const int tid = threadIdx.x;

    // Stage both weight matrices into LDS (padded rows).
    for (int i = tid; i < 64 * 64; i += 128) {
        const int rr = i >> 6, cc = i & 63;
        sWm[rr * WS + cc] = Wm[i];
        sWe[rr * WS + cc] = We[i];
    }
    __syncthreads();

    const int wave = tid >> 5;
    const int lane = tid & 31;
    const int r    = lane & 15;   // A row (per tile) / B,C,D column
    const int hf   = lane >> 4;   // half-wave selector per ISA layouts
    const int row0 = (blockIdx.x * 4 + wave) * (16 * R);  // first of this wave's 16*R rows

    // ---- Gather + sum into A fragments for GEMM1 (x_1st is L2-resident: 10 MB) ----
    v2f a1[R][16];
#pragma unroll
    for (int t = 0; t < R; ++t) {
        const int grow = row0 + 16 * t + r;
        const int b    = grow / (kM * 4);
        const int rem  = grow - b * (kM * 4);
        const int m    = rem >> 2;
        const int c    = rem & 3;
        const int e0   = edge[m];        // edge[0][m]
        const int e1   = edge[kM + m];   // edge[1][m]
        // NODE_I[c] = c>>1 (0,0,1,1); NODE_J[c] = c&1 (0,1,0,1)
        const v2f* pi = (const v2f*)(x1 + (((size_t)b * kT + e0) * 2 + (c >> 1)) * kD);
        const v2f* pj = (const v2f*)(x1 + (((size_t)b * kT + e1) * 2 + (c & 1)) * kD);
#pragma unroll
        for (int k = 0; k < 16; ++k)
            a1[t][k] = pi[2 * k + hf] + pj[2 * k + hf];
    }

    float* hb = &sH[wave][0];

    // ---- GEMM1: mlp = s @ Wm^T + bm -> LDS (row-major, padded) ----
    // (x_2nd is folded in later, in GEMM2's A-fragment domain: linearity of GEMM2.)
#pragma unroll
    for (int nt = 0; nt < 4; ++nt) {
        v8f acc[R];
#pragma unroll
        for (int t = 0; t < R; ++t)
            acc[t] = v8f{0.f, 0.f, 0.f, 0.f, 0.f, 0.f, 0.f, 0.f};
#pragma unroll
        for (int k = 0; k < 16; ++k) {
            // B fragment (4x16): { Wm[n][4k+2hf], Wm[n][4k+2hf+1] }, n = nt*16 + r
            const v2f bf = *(const v2f*)&sWm[(nt * 16 + r) * WS + 4 * k + 2 * hf];
#pragma unroll
            for (int t = 0; t < R; ++t)
                acc[t] = __builtin_amdgcn_wmma_f32_16x16x4_f32(
                    false, a1[t][k], false, bf, (short)0, acc[t], false, false);
        }
        const int   col  = nt * 16 + r;     // N index held by this lane (C/D layout)
        const float bias = bm[col];
#pragma unroll
        for (int t = 0; t < R; ++t) {
#pragma unroll
            for (int v = 0; v < 8; ++v) {
                const int lrow = 16 * t + v + 8 * hf;  // wave-local row of acc[t][v]
                hb[lrow * WS + col] = acc[t][v] + bias;
            }
        }
    }
    __syncthreads();  // order per-wave h writes vs cross-lane reads

    // A fragments for GEMM2: h = mlp + bm (from LDS, conflict-free) + x_2nd (global, v2f
    // in A-layout; consecutive k steps consume full 64B lines via L0/L2).
    v2f a2[R][16];
#pragma unroll
    for (int t = 0; t < R; ++t) {
        const v2f* px2 = (const v2f*)(x2 + (size_t)(row0 + 16 * t + r) * kD);
#pragma unroll
        for (int k = 0; k < 16; ++k)
            a2[t][k] = *(const v2f*)&hb[(16 * t + r) * WS + 4 * k + 2 * hf]
                     + px2[2 * k + hf];
    }

    // ---- GEMM2: out = h @ We^T + be ----
#pragma unroll
    for (int nt = 0; nt < 4; ++nt) {
        v8f acc[R];
#pragma unroll
        for (int t = 0; t < R; ++t)
            acc[t] = v8f{0.f, 0.f, 0.f, 0.f, 0.f, 0.f, 0.f, 0.f};
#pragma unroll
        for (int k = 0; k < 16; ++k) {
            const v2f bf = *(const v2f*)&sWe[(nt * 16 + r) * WS + 4 * k + 2 * hf];
#pragma unroll
            for (int t = 0; t < R; ++t)
                acc[t] = __builtin_amdgcn_wmma_f32_16x16x4_f32(
                    false, a2[t][k], false, bf, (short)0, acc[t], false, false);
        }
        const int   col  = nt * 16 + r;
        const float bias = be[col];
#pragma unroll
        for (int t = 0; t < R; ++t) {
#pragma unroll
            for (int v = 0; v < 8; ++v) {
                const int lrow = 16 * t + v + 8 * hf;
                out[(size_t)(row0 + lrow) * kD + col] = acc[t][v] + bias;
            }
        }
    }
}

extern "C" void kernel_launch(void* const* d_in, const int* in_sizes, int n_in,
                              void* d_out, int out_size, void* d_ws, size_t ws_size,
                              hipStream_t stream) {
    (void)in_sizes; (void)n_in; (void)out_size; (void)d_ws; (void)ws_size;
    const float* x1   = (const float*)d_in[0];
    const float* x2   = (const float*)d_in[1];
    const float* Wm   = (const float*)d_in[2];
    const float* bm   = (const float*)d_in[3];
    const float* We   = (const float*)d_in[4];
    const float* be   = (const float*)d_in[5];
    const int*   edge = (const int*)  d_in[6];
    float*       out  = (float*)d_out;

    // 800000 rows total, 16*R rows per wave, 4 waves per 128-thread block.
    const int nblocks = (kB * kM * 4) / (16 * R * 4);  // 6250
    gnn_fused_wmma_f32<<<nblocks, 128, 0, stream>>>(x1, x2, Wm, bm, We, be, edge, out);
}